// GCN_73512660238643
// MI455X (gfx1250) — compile-verified
//
#include <hip/hip_runtime.h>

typedef __attribute__((ext_vector_type(2))) float v2f;
typedef __attribute__((ext_vector_type(8))) float v8f;

#define IN_C  256
#define HID_C 128
#define OUT_C 64

// ---------------- degree / normalization ----------------

__global__ void k_init_deg(float* __restrict__ deg, int n) {
  int i = blockIdx.x * blockDim.x + threadIdx.x;
  if (i < n) deg[i] = 1.0f;                       // self-loop contributes 1
}

__global__ void k_deg_scatter(const long long* __restrict__ dst,
                              float* __restrict__ deg, int E) {
  int e = blockIdx.x * blockDim.x + threadIdx.x;
  if (e < E) atomicAdd(&deg[(int)dst[e]], 1.0f);
}

__global__ void k_rsqrt_inplace(float* __restrict__ d, int n) {
  int i = blockIdx.x * blockDim.x + threadIdx.x;
  if (i < n) d[i] = rsqrtf(d[i]);                 // deg >= 1 always (self loop)
}

// ---------------- fp32 WMMA GEMM: C[M,NC] = op(A[M,K]) @ B[K,NC] ----------------
// One wave32 computes one 16x16 tile via V_WMMA_F32_16X16X4_F32 (K-step 4).
// A layout (ISA 7.12.2, 32-bit A 16x4): lane L holds row M=L%16,
//   VGPR0 = A[M][kk + 2*(L/16)], VGPR1 = A[M][kk + 2*(L/16) + 1].
// B (4x16) mirrors A transposed; C/D: acc[r] -> row r + 8*(L/16), col L%16.

template<int K, int NC, bool RELU>
__global__ void k_gemm_wmma(const float* __restrict__ A, const float* __restrict__ B,
                            float* __restrict__ C) {
  const int lane = threadIdx.x;                               // 0..31 (wave32)
  const int tm   = blockIdx.x;                                // 16-row tile
  const int tn   = blockIdx.y * blockDim.y + threadIdx.y;     // 16-col tile
  const int rowA = lane & 15;
  const int kh   = (lane >> 4) << 1;                          // 0 or 2
  const int col  = tn * 16 + (lane & 15);

  const float* __restrict__ arow = A + (size_t)(tm * 16 + rowA) * K;

  v8f acc = {};
#pragma unroll 4
  for (int kk = 0; kk < K; kk += 4) {
    float a0 = arow[kk + kh];
    float a1 = arow[kk + kh + 1];
    if (RELU) { a0 = fmaxf(a0, 0.0f); a1 = fmaxf(a1, 0.0f); }  // fused relu(y1)
    v2f a; a[0] = a0; a[1] = a1;
    v2f b; b[0] = B[(size_t)(kk + kh)     * NC + col];
           b[1] = B[(size_t)(kk + kh + 1) * NC + col];
    acc = __builtin_amdgcn_wmma_f32_16x16x4_f32(
        /*neg_a=*/false, a, /*neg_b=*/false, b,
        /*c_mod=*/(short)0, acc, /*reuse_a=*/false, /*reuse_b=*/false);
  }

  const int r0 = (lane >> 4) << 3;
#pragma unroll
  for (int r = 0; r < 8; ++r)
    C[(size_t)(tm * 16 + r0 + r) * NC + col] = acc[r];
}

// ---------------- aggregation: out = D^-1/2 (A+I) D^-1/2 h + bias ----------------

// init with self-loop term + bias: out[i][c] = h[i][c]*dis[i]^2 + bias[c]
template<int C>
__global__ void k_agg_init(const float* __restrict__ h, const float* __restrict__ dis,
                           const float* __restrict__ bias, float* __restrict__ out, int n) {
  int idx = blockIdx.x * blockDim.x + threadIdx.x;
  if (idx < n * C) {
    int i = idx / C, c = idx - i * C;
    float s = dis[i];
    out[idx] = h[idx] * (s * s) + bias[c];
  }
}

// one wave32 per edge; each lane handles C/32 contiguous channels.
// scatter-adds resolve at the L2 atomic units; h and out both fit in 192MB L2.
template<int C>
__global__ void k_agg_scatter(const float* __restrict__ h, const float* __restrict__ dis,
                              const long long* __restrict__ src,
                              const long long* __restrict__ dst,
                              float* __restrict__ out, int E) {
  int gtid = blockIdx.x * blockDim.x + threadIdx.x;
  int e    = gtid >> 5;
  int lane = threadIdx.x & 31;
  if (e >= E) return;
  int s = (int)src[e];
  int d = (int)dst[e];
  float w = dis[s] * dis[d];
  constexpr int V = C / 32;
  const float* __restrict__ hs = h   + (size_t)s * C + lane * V;
  float*       __restrict__ od = out + (size_t)d * C + lane * V;
#pragma unroll
  for (int v = 0; v < V; ++v)
    atomicAdd(&od[v], hs[v] * w);
}

// ---------------- driver ----------------

extern "C" void kernel_launch(void* const* d_in, const int* in_sizes, int n_in,
                              void* d_out, int out_size, void* d_ws, size_t ws_size,
                              hipStream_t stream) {
  const float*     X   = (const float*)d_in[0];        // [N, 256]
  const long long* EI  = (const long long*)d_in[1];    // [2, E] int64
  const float*     W1  = (const float*)d_in[2];        // [256, 128]
  const float*     b1  = (const float*)d_in[3];        // [128]
  const float*     W2  = (const float*)d_in[4];        // [128, 64]
  const float*     b2  = (const float*)d_in[5];        // [64]
  float*           out = (float*)d_out;                // [N, 64]

  const int N = in_sizes[0] / IN_C;                    // 50000 (multiple of 16)
  const int E = in_sizes[1] / 2;                       // 500000
  const long long* src = EI;
  const long long* dst = EI + E;

  // workspace layout (peak ~51.5 MB): dis | h1 | y1 ; h2 aliases h1 (dead by then)
  char*  ws  = (char*)d_ws;
  float* dis = (float*)ws;
  size_t off = ((size_t)N * 4 + 255) & ~(size_t)255;
  float* h1  = (float*)(ws + off);
  float* y1  = h1 + (size_t)N * HID_C;
  float* h2  = h1;

  // normalization coefficients
  k_init_deg   <<<(N + 255) / 256, 256, 0, stream>>>(dis, N);
  k_deg_scatter<<<(E + 255) / 256, 256, 0, stream>>>(dst, dis, E);
  k_rsqrt_inplace<<<(N + 255) / 256, 256, 0, stream>>>(dis, N);

  dim3 gblk(32, 4);                                    // 4 wave32 per block

  // layer 1: h1 = X @ W1 ; y1 = norm-aggregate(h1) + b1
  k_gemm_wmma<IN_C, HID_C, false><<<dim3(N / 16, HID_C / 64), gblk, 0, stream>>>(X, W1, h1);
  k_agg_init<HID_C><<<((size_t)N * HID_C + 255) / 256, 256, 0, stream>>>(h1, dis, b1, y1, N);
  k_agg_scatter<HID_C><<<(E + 7) / 8, 256, 0, stream>>>(h1, dis, src, dst, y1, E);

  // layer 2: h2 = relu(y1) @ W2 (relu fused into A load) ; out = aggregate(h2) + b2
  k_gemm_wmma<HID_C, OUT_C, true><<<dim3(N / 16, OUT_C / 64), gblk, 0, stream>>>(y1, W2, h2);
  k_agg_init<OUT_C><<<((size_t)N * OUT_C + 255) / 256, 256, 0, stream>>>(h2, dis, b2, out, N);
  k_agg_scatter<OUT_C><<<(E + 7) / 8, 256, 0, stream>>>(h2, dis, src, dst, out, E);
}